// DAEL_48868137894562
// MI455X (gfx1250) — compile-verified
//
#include <hip/hip_runtime.h>

typedef float v2f  __attribute__((ext_vector_type(2)));
typedef float v8f  __attribute__((ext_vector_type(8)));
typedef unsigned int u32x4 __attribute__((ext_vector_type(4)));
typedef int   i32x4 __attribute__((ext_vector_type(4)));
typedef int   i32x8 __attribute__((ext_vector_type(8)));

#define N_EXPERTS 8
#define FDIM1 1024
#define FDIM2 512
#define NCLS 64
#define KC 128                     // K-chunk of x staged in LDS
#define NCHUNK (FDIM1 / KC)        // 8
#define XS_STRIDE 132              // 128 + 4 pad -> conflict-free A-frag reads
#define XS_BUF_FLOATS (16 * XS_STRIDE)
#define HS_STRIDE 516              // 512 + 4 pad
#define LG_STRIDE 68               // 64 + 4 pad

// workspace int layout
#define WS_COUNTS 0
#define WS_PADOFF 16               // 9 entries (exclusive padded prefix, [8] = total)
#define WS_RUN    32               // 8 running cursors
#define WS_PERM   64               // padded permutation (row index or -1)

#if __has_builtin(__builtin_amdgcn_tensor_load_to_lds) && __has_builtin(__builtin_amdgcn_s_wait_tensorcnt)
#define HAVE_TDM 1
#else
#define HAVE_TDM 0
#endif

__global__ void moe_init(int* wsi, int total) {
    int i = blockIdx.x * blockDim.x + threadIdx.x;
    if (i >= total) return;
    wsi[i] = (i < WS_PERM) ? 0 : -1;
}

__global__ void moe_hist(const int* __restrict__ dom, int* wsi, int n) {
    int i = blockIdx.x * blockDim.x + threadIdx.x;
    if (i < n) atomicAdd(&wsi[WS_COUNTS + dom[i]], 1);
}

__global__ void moe_scan(int* wsi) {
    if (threadIdx.x == 0 && blockIdx.x == 0) {
        int off = 0;
        for (int e = 0; e < N_EXPERTS; ++e) {
            wsi[WS_PADOFF + e] = off;
            wsi[WS_RUN + e]    = off;
            int c = wsi[WS_COUNTS + e];
            off += (c + 15) & ~15;           // pad each expert segment to 16 rows
        }
        wsi[WS_PADOFF + N_EXPERTS] = off;
    }
}

__global__ void moe_scatter(const int* __restrict__ dom, int* wsi, int n) {
    int i = blockIdx.x * blockDim.x + threadIdx.x;
    if (i < n) {
        int e = dom[i];
        int pos = atomicAdd(&wsi[WS_RUN + e], 1);
        wsi[WS_PERM + pos] = i;
    }
}

// One block per 16-row single-expert tile. 256 threads = 8 wave32.
__global__ __launch_bounds__(256) void moe_main(
    const float* __restrict__ x, const float* __restrict__ W1,
    const float* __restrict__ b1, const float* __restrict__ W2,
    const float* __restrict__ b2, const int* __restrict__ wsi,
    float* __restrict__ out, int nrows)
{
    __shared__ float xs[2][XS_BUF_FLOATS];  // 16.9 KB: double-buffered x chunk [16][128]
    __shared__ float hs[16 * HS_STRIDE];    // 33.0 KB: hidden [16][512]
    __shared__ float lg[16 * LG_STRIDE];    //  4.4 KB: logits [16][64]
    __shared__ int   rows[16];

    const int base = blockIdx.x * 16;
    const int* padoff = wsi + WS_PADOFF;
    if (base >= padoff[N_EXPERTS]) return;           // block-uniform exit
    int e = 0;
    #pragma unroll
    for (int q = 1; q < N_EXPERTS; ++q) e += (base >= padoff[q]) ? 1 : 0;

    const int tid  = threadIdx.x;
    const int wave = tid >> 5;       // 0..7
    const int lane = tid & 31;
    const int half = lane >> 4;      // A/B frag: upper half-wave handles K+2,K+3
    const int l16  = lane & 15;

    if (tid < 16) rows[tid] = wsi[WS_PERM + base + tid];
    __syncthreads();

#if HAVE_TDM
    // ---- TDM gather descriptor (uniform): pull 16 permuted rows -> LDS ----
    // Valid rows are a prefix of the tile (scatter fills segments densely).
    int cnt_e = wsi[WS_COUNTS + e];
    int valid = cnt_e - (base - padoff[e]);
    if (valid > 16) valid = 16;
    int idxw[8];
    #pragma unroll
    for (int j = 0; j < 8; ++j) {
        int r0 = rows[2 * j];     r0 = r0 < 0 ? 0 : r0;
        int r1 = rows[2 * j + 1]; r1 = r1 < 0 ? 0 : r1;
        idxw[j] = __builtin_amdgcn_readfirstlane((r0 & 0xFFFF) | (r1 << 16));
    }
    i32x8 g1;
    g1[0] = (2 << 16)      // data_size = 4B
          | (1 << 20)      // pad_enable
          | (6 << 22)      // pad_interval = 128 DWORDs (one row)
          | (3 << 25);     // pad_amount   = 4 DWORDs  -> LDS stride 132
    g1[1] = (int)(((unsigned)FDIM1 & 0xFFFFu) << 16);                            // tensor_dim0 lo
    g1[2] = (int)(((unsigned)FDIM1 >> 16) | (((unsigned)nrows & 0xFFFFu) << 16));// dim0 hi | dim1 lo
    g1[3] = (int)(((unsigned)nrows >> 16) | ((unsigned)KC << 16));               // dim1 hi | tile_dim0
    g1[4] = valid;                                                               // tile_dim1 = #indices
    g1[5] = FDIM1;                                                               // tensor_dim0_stride lo
    g1[6] = 0;
    g1[7] = 0;
    i32x4 g2; g2[0] = idxw[0]; g2[1] = idxw[1]; g2[2] = idxw[2]; g2[3] = idxw[3];
    i32x4 g3; g3[0] = idxw[4]; g3[1] = idxw[5]; g3[2] = idxw[6]; g3[3] = idxw[7];
    const unsigned xs_lds0 = (unsigned)(size_t)&xs[0][0];  // flat addr low 32 = LDS byte offset
    const unsigned long long xg = (unsigned long long)(size_t)x;

    auto tdm_issue = [&](int chunk) {
        unsigned long long ga = xg + (unsigned long long)(chunk * KC) * 4ull;
        u32x4 g0;
        g0[0] = 0x80000001u;   // count=1 | gather_mode | 16-bit indices
        g0[1] = xs_lds0 + (unsigned)((chunk & 1) * (XS_BUF_FLOATS * 4));
        g0[2] = (unsigned)ga;
        g0[3] = (unsigned)((ga >> 32) & 0x01FFFFFFu) | (2u << 30);  // type=2 (image)
#if defined(__clang_major__) && (__clang_major__ >= 23)
        i32x8 gz = {};
        __builtin_amdgcn_tensor_load_to_lds(g0, g1, g2, g3, gz, 0);
#else
        __builtin_amdgcn_tensor_load_to_lds(g0, g1, g2, g3, 0);
#endif
    };
    if (wave == 0) tdm_issue(0);
#endif

    // ---- Layer 1: h[16][512] = relu(x[16][1024] * W1[e] + b1[e]) ----
    const float* W1e = W1 + (long)e * (FDIM1 * FDIM2);
    v8f acc[4] = {};                 // wave covers columns [wave*64, wave*64+64)

    for (int c = 0; c < NCHUNK; ++c) {
#if HAVE_TDM
        if (wave == 0) __builtin_amdgcn_s_wait_tensorcnt((short)0);  // chunk c resident
        __syncthreads();                                             // publish to all waves
        if (wave == 0 && c + 1 < NCHUNK) tdm_issue(c + 1);           // overlap next chunk
#else
        {   // fallback: coalesced float4 gather into buf c&1 (pad rows -> 0)
            const int ldr = tid >> 4, ldc = tid & 15;
            const int myrow = rows[ldr];
            const float* xsrc = x + (long)myrow * FDIM1 + c * KC;
            float* dst = &xs[c & 1][ldr * XS_STRIDE];
            #pragma unroll
            for (int j = 0; j < KC / 64; ++j) {
                int cc = (ldc + j * 16) * 4;
                float4 v = make_float4(0.f, 0.f, 0.f, 0.f);
                if (myrow >= 0) v = *(const float4*)(xsrc + cc);
                *(float4*)(dst + cc) = v;
            }
        }
        __syncthreads();   // single barrier/chunk: next stage hits the other buffer
#endif
        const float* arow = &xs[c & 1][l16 * XS_STRIDE];
        const int kc = c * KC;
        for (int k0 = 0; k0 < KC; k0 += 4) {
            int ksl = k0 + 2 * half;
            v2f a = *(const v2f*)(arow + ksl);            // ds_load b64, conflict-free
            const float* bbase = W1e + (long)(kc + ksl) * FDIM2 + wave * 64 + l16;
            #pragma unroll
            for (int t = 0; t < 4; ++t) {
                const float* bp = bbase + t * 16;
                v2f b; b.x = bp[0]; b.y = bp[FDIM2];      // B[k][n], B[k+1][n] (L2-resident)
                acc[t] = __builtin_amdgcn_wmma_f32_16x16x4_f32(
                    false, a, false, b, (short)0, acc[t], false, false);
            }
        }
    }

    // bias + ReLU -> hs
    {
        const float* b1e = b1 + e * FDIM2;
        #pragma unroll
        for (int t = 0; t < 4; ++t) {
            int n = wave * 64 + t * 16 + l16;
            float bias = b1e[n];
            #pragma unroll
            for (int r = 0; r < 8; ++r) {
                int m = r + 8 * half;                     // C/D layout: VGPR r -> row r / r+8
                float v = acc[t][r] + bias;
                hs[m * HS_STRIDE + n] = fmaxf(v, 0.f);
            }
        }
    }
    __syncthreads();

    // ---- Layer 2: logits[16][64] = h * W2[e] + b2[e] (waves 0..3, wave-uniform) ----
    if (wave < 4) {
        const float* W2e = W2 + (long)e * (FDIM2 * NCLS);
        v8f acc2 = {};
        const float* hrow = hs + l16 * HS_STRIDE;
        int ncol = wave * 16 + l16;
        for (int k0 = 0; k0 < FDIM2; k0 += 4) {
            int ks = k0 + 2 * half;
            v2f a = *(const v2f*)(hrow + ks);
            const float* bp = W2e + (long)ks * NCLS + ncol;
            v2f b; b.x = bp[0]; b.y = bp[NCLS];
            acc2 = __builtin_amdgcn_wmma_f32_16x16x4_f32(
                false, a, false, b, (short)0, acc2, false, false);
        }
        float bias = b2[e * NCLS + ncol];
        #pragma unroll
        for (int r = 0; r < 8; ++r) {
            int m = r + 8 * half;
            lg[m * LG_STRIDE + ncol] = acc2[r] + bias;
        }
    }
    __syncthreads();

    // ---- softmax over 64 classes, one thread per row ----
    if (tid < 16) {
        float* lr = lg + tid * LG_STRIDE;
        float mx = lr[0];
        for (int c = 1; c < NCLS; ++c) mx = fmaxf(mx, lr[c]);
        float s = 0.f;
        for (int c = 0; c < NCLS; ++c) { float ev = __expf(lr[c] - mx); lr[c] = ev; s += ev; }
        float inv = 1.f / s;
        for (int c = 0; c < NCLS; ++c) lr[c] *= inv;
    }
    __syncthreads();

    // ---- scattered store (skip pad rows) ----
    {
        int r = tid >> 4;
        int row = rows[r];
        if (row >= 0) {
            int c = (tid & 15) * 4;
            float4 v = *(float4*)(&lg[r * LG_STRIDE + c]);
            *(float4*)(out + (long)row * NCLS + c) = v;
        }
    }
}

extern "C" void kernel_launch(void* const* d_in, const int* in_sizes, int n_in,
                              void* d_out, int out_size, void* d_ws, size_t ws_size,
                              hipStream_t stream) {
    const int*   domain = (const int*)  d_in[0];
    const float* x      = (const float*)d_in[1];
    const float* W1     = (const float*)d_in[2];
    const float* b1     = (const float*)d_in[3];
    const float* W2     = (const float*)d_in[4];
    const float* b2     = (const float*)d_in[5];
    float* out = (float*)d_out;
    int*   wsi = (int*)d_ws;

    const int n = in_sizes[0];                                   // 16384
    const int max_tiles  = (n + 15 * N_EXPERTS + 15) / 16;       // 1032
    const int init_total = WS_PERM + max_tiles * 16;

    moe_init   <<<(init_total + 255) / 256, 256, 0, stream>>>(wsi, init_total);
    moe_hist   <<<(n + 255) / 256,          256, 0, stream>>>(domain, wsi, n);
    moe_scan   <<<1, 32, 0, stream>>>(wsi);
    moe_scatter<<<(n + 255) / 256,          256, 0, stream>>>(domain, wsi, n);
    moe_main   <<<max_tiles, 256, 0, stream>>>(x, W1, b1, W2, b2, wsi, out, n);
}